// IntraSampleNTXEntLoss_85864986182122
// MI455X (gfx1250) — compile-verified
//
#include <hip/hip_runtime.h>
#include <math.h>

// NT-Xent loss for inp (C=8, V=2, B=4096, D=512) f32 -> scalar f32.
// Per batch element: 16x16 Gram via V_WMMA_F32_16X16X4_F32 chained over K=512,
// normalization from the Gram diagonal, per-crop logsumexp, atomic mean.

typedef __attribute__((ext_vector_type(2))) float v2f;
typedef __attribute__((ext_vector_type(8))) float v8f;

#define C_CROPS   8
#define V_VIEWS   2
#define BATCH     4096
#define DIM       512
#define NVEC      16          // C*V vectors per batch element
#define TEMP_INV  10.0f       // 1 / 0.1
#define NORM_EPSF 1e-12f

__global__ void ntxent_zero_kernel(float* __restrict__ out) {
    out[0] = 0.0f;
}

__global__ __launch_bounds__(256) void
ntxent_gram_wmma_kernel(const float* __restrict__ inp, float* __restrict__ out) {
    const int lane = threadIdx.x & 31;   // wave32
    const int wave = threadIdx.x >> 5;   // 8 waves per block
    const int b    = blockIdx.x * 8 + wave;

    const int vec  = lane & 15;          // which of the 16 (crop,view) vectors
    const int half = lane >> 4;          // K sub-offset selector

    // Per-wave 16x16 Gram scratch in LDS (8 waves * 256 f32 = 8 KB).
    __shared__ float lds[8][NVEC * NVEC];
    float* g = lds[wave];

    // Memory layout (C,V,B,D): vector p=(c*V+v) lives at p*B*D + b*D.
    // Lane L loads X[vec][4k + 2*half + {0,1}] each step: exactly the
    // 16x4 f32 A-operand layout; B = A^T (Gram) has identical lane contents,
    // so the same register pair feeds both WMMA sources.
    const float* ptr = inp + (size_t)vec * (size_t)(BATCH * DIM)
                           + (size_t)b * (size_t)DIM
                           + (size_t)(2 * half);

    v8f acc = {};   // 16x16 f32 accumulator (C/D layout: VGPR r -> rows r, r+8)
    #pragma unroll 4
    for (int k = 0; k < DIM / 4; ++k) {
        v2f a = *reinterpret_cast<const v2f*>(ptr + 4 * k);   // global_load_b64
        acc = __builtin_amdgcn_wmma_f32_16x16x4_f32(
                  /*neg_a=*/false, a, /*neg_b=*/false, a,
                  /*c_mod=*/(short)0, acc,
                  /*reuse_a=*/false, /*reuse_b=*/false);
    }

    // Spill Gram to LDS: VGPR r holds G[r + 8*half][vec].
    #pragma unroll
    for (int r = 0; r < 8; ++r) {
        g[(r + 8 * half) * NVEC + vec] = acc[r];
    }
    __syncthreads();

    // Lanes 0..7: one crop each. Row q = 2i (view 0 is the query view).
    float loss = 0.0f;
    if (lane < C_CROPS) {
        const int i = lane;
        const int q = 2 * i;
        const float rq = 1.0f / fmaxf(sqrtf(g[q * NVEC + q]), NORM_EPSF);

        float s[NVEC];
        #pragma unroll
        for (int p = 0; p < NVEC; ++p) {
            const float rp = 1.0f / fmaxf(sqrtf(g[p * NVEC + p]), NORM_EPSF);
            s[p] = g[q * NVEC + p] * rq * rp * TEMP_INV;
        }

        const float pos = s[q + 1];                 // sim_all[i,b,i,1]
        // logits = {pos} U {s[p] : p's crop != i}  (same-crop entries -> -inf)
        float m = pos;
        #pragma unroll
        for (int p = 0; p < NVEC; ++p)
            if ((p >> 1) != i) m = fmaxf(m, s[p]);
        float sum = __expf(pos - m);
        #pragma unroll
        for (int p = 0; p < NVEC; ++p)
            if ((p >> 1) != i) sum += __expf(s[p] - m);

        loss = m + __logf(sum) - pos;               // lse - sim_pos
    }

    // Sum the 8 crop losses of this wave's batch element (lanes 8..31 hold 0).
    loss += __shfl_down(loss, 4, 32);
    loss += __shfl_down(loss, 2, 32);
    loss += __shfl_down(loss, 1, 32);

    if (lane == 0) {
        atomicAdd(out, loss * (1.0f / (float)(C_CROPS * BATCH)));
    }
}

extern "C" void kernel_launch(void* const* d_in, const int* in_sizes, int n_in,
                              void* d_out, int out_size, void* d_ws, size_t ws_size,
                              hipStream_t stream) {
    (void)in_sizes; (void)n_in; (void)out_size; (void)d_ws; (void)ws_size;
    const float* inp = (const float*)d_in[0];
    float* out       = (float*)d_out;

    ntxent_zero_kernel<<<1, 1, 0, stream>>>(out);
    ntxent_gram_wmma_kernel<<<BATCH / 8, 256, 0, stream>>>(inp, out);
}